// BehlerSymmertryFunctions_40243843564182
// MI455X (gfx1250) — compile-verified
//
#include <hip/hip_runtime.h>

// Behler symmetry functions (G2 + G4) for MI455X / gfx1250.
// One wave32 per atom. The 32x32 pairwise dot-product (Gram) matrix of
// neighbor displacement vectors is computed with 4x V_WMMA_F32_16X16X4_F32
// (fp32 WMMA: cos and rjk^2 are catastrophic-cancellation sensitive).
// cos_ijk = dot/(ri*rk), rjk^2 = ri^2 + rk^2 - 2*dot.
// Reductions use an LDS transpose; outputs are written coalesced.
// Transcendentals use raw v_exp_f32 / v_cos_f32 with scale constants folded.

#define N_G2   8
#define N_G4   43
#define N_FEAT 51
#define KNEI   32
#define RSTRIDE 36            // floats per reduction row: 144B = 16B aligned
#define RCUTF  8.0f
#define RCUTSQ 64.0f

typedef float v2f __attribute__((ext_vector_type(2)));
typedef float v8f __attribute__((ext_vector_type(8)));

// Same-wave LDS ordering is guaranteed by hardware (DS ops execute in issue
// order per wave); we only need to stop the compiler from reordering.
__device__ __forceinline__ void wave_lds_fence() {
#if __has_builtin(__builtin_amdgcn_wave_barrier)
  __builtin_amdgcn_wave_barrier();
#endif
  asm volatile("" ::: "memory");
}

// exp(x) with pre-scaled argument: caller passes x already in log2 units.
__device__ __forceinline__ float exp2_raw(float x_log2) {
#if __has_builtin(__builtin_amdgcn_exp2f)
  return __builtin_amdgcn_exp2f(x_log2);     // raw v_exp_f32 (base 2)
#else
  return __expf(x_log2 * 0.6931471805599453f);
#endif
}

// cos(2*pi*x) via raw v_cos_f32 (hardware input is in revolutions).
__device__ __forceinline__ float cos_rev(float x_rev) {
#if __has_builtin(__builtin_amdgcn_cosf)
  return __builtin_amdgcn_cosf(x_rev);
#else
  return __cosf(x_rev * 6.283185307179586f);
#endif
}

__device__ __forceinline__ float fc_cut(float r) {
  // 0.5*(cos(pi*r/RCUT)+1) = 0.5*(cos(2*pi * r/16)+1) for r < RCUT, else 0
  float v = 0.5f * (cos_rev(r * 0.0625f) + 1.0f);
  return (r < RCUTF) ? v : 0.0f;
}

#define LOG2E 1.4426950408889634f

__global__ __launch_bounds__(256) void behler_g2g4_kernel(
    const float* __restrict__ pos,
    const int*   __restrict__ numnei,
    const int*   __restrict__ neighs,
    float*       __restrict__ out,
    int n_atoms)
{
  const int lane = threadIdx.x & 31;
  const int wv   = threadIdx.x >> 5;          // 8 waves / block
  const int atom = blockIdx.x * 8 + wv;
  if (atom >= n_atoms) return;                // wave-uniform exit

  // Per-wave staging: displacement rows (x,y,z,0) and (r, fc*valid)
  __shared__ float ldsR[8][KNEI][4];
  __shared__ float ldsS[8][KNEI][2];
  // Per-wave reduction tile: [feature][lane], padded rows
  __shared__ float red[8][N_FEAT][RSTRIDE];

  // G2 etas and the 7 distinct G4 etas, pre-multiplied by -log2(e) so each
  // exponential is a single v_mul + v_exp_f32.
  const float G2E_L2[N_G2] = {-0.0036f * LOG2E, -0.036f * LOG2E,
                              -0.071f  * LOG2E, -0.125f * LOG2E,
                              -0.214f  * LOG2E, -0.357f * LOG2E,
                              -0.714f  * LOG2E, -1.428f * LOG2E};
  const float ETA7_L2[7]   = {-0.0001f * LOG2E, -0.003f * LOG2E,
                              -0.008f  * LOG2E, -0.015f * LOG2E,
                              -0.025f  * LOG2E, -0.045f * LOG2E,
                              -0.08f   * LOG2E};

  // ---- Per-neighbor geometry ----
  const float px = pos[atom * 3 + 0];
  const float py = pos[atom * 3 + 1];
  const float pz = pos[atom * 3 + 2];
  const int nb = neighs[atom * KNEI + lane];
  const float x = pos[nb * 3 + 0] - px;
  const float y = pos[nb * 3 + 1] - py;
  const float z = pos[nb * 3 + 2] - pz;
  const float rsq = x * x + y * y + z * z;
  const float r   = sqrtf(rsq);
  const int   nn  = numnei[atom];
  const float valid = (lane < nn && rsq <= RCUTSQ) ? 1.0f : 0.0f;
  const float fcm = fc_cut(r) * valid;        // fc * mask_j folded together

  ldsR[wv][lane][0] = x;
  ldsR[wv][lane][1] = y;
  ldsR[wv][lane][2] = z;
  ldsR[wv][lane][3] = 0.0f;
  ldsS[wv][lane][0] = r;
  ldsS[wv][lane][1] = fcm;

  wave_lds_fence();   // intra-wave sharing only; HW keeps same-wave DS in order

  // ---- G2: 8 radial features (RS == 0) ----
  float g2v[N_G2];
#pragma unroll
  for (int e = 0; e < N_G2; ++e)
    g2v[e] = exp2_raw(G2E_L2[e] * rsq) * fcm;

  // ---- Build WMMA fragments for R (32x4) ----
  // A 16x4 f32 layout: lanes 0-15 hold M=lane, K0/K1 in VGPR0/1;
  // lanes 16-31 hold M=lane-16, K2/K3. B (4x16) mirrors it with N=lane&15,
  // so one fragment loader serves both operands.
  const int lhalf = lane & 15;
  const int hi    = lane >> 4;    // 0 or 1
  const int kb    = hi * 2;       // K base 0 or 2
  v2f f0, f1;
  f0.x = ldsR[wv][lhalf][kb];       f0.y = ldsR[wv][lhalf][kb + 1];
  f1.x = ldsR[wv][16 + lhalf][kb];  f1.y = ldsR[wv][16 + lhalf][kb + 1];

  // Per-lane j-row data (16 rows this lane owns in the C layout)
  float rj[16], fj[16];
#pragma unroll
  for (int mi = 0; mi < 2; ++mi) {
#pragma unroll
    for (int v = 0; v < 8; ++v) {
      const int j = mi * 16 + hi * 8 + v;
      rj[mi * 8 + v] = ldsS[wv][j][0];
      fj[mi * 8 + v] = ldsS[wv][j][1];
    }
  }

  float acc[N_G4];
#pragma unroll
  for (int p = 0; p < N_G4; ++p) acc[p] = 0.0f;

  // ---- G4 over all ordered pairs (j,k), weight 0.5, diagonal masked ----
#pragma unroll
  for (int ni = 0; ni < 2; ++ni) {
    const int k = ni * 16 + lhalf;           // this lane's column
    const float rk   = ldsS[wv][k][0];
    const float fk   = ldsS[wv][k][1];
    const float rksq = rk * rk;
    const v2f fb = (ni == 0) ? f0 : f1;

#pragma unroll
    for (int mi = 0; mi < 2; ++mi) {
      const v2f fa = (mi == 0) ? f0 : f1;
      v8f cz = {};
      // D = A x B : 16x16 tile of the Gram matrix, fp32 accumulate
      v8f D = __builtin_amdgcn_wmma_f32_16x16x4_f32(
          /*neg_a=*/false, fa, /*neg_b=*/false, fb,
          /*c_mod=*/(short)0, cz, /*reuse_a=*/false, /*reuse_b=*/false);

#pragma unroll
      for (int v = 0; v < 8; ++v) {
        const int j = mi * 16 + hi * 8 + v;
        const float dot  = D[v];
        const float rjv  = rj[mi * 8 + v];
        const float fjv  = fj[mi * 8 + v];
        const float rjsq = rjv * rjv;

        const float rjk_sq = fmaxf(rjsq + rksq - 2.0f * dot, 0.0f);
        const float rjk    = sqrtf(rjk_sq);
        const float cosv   = dot * __builtin_amdgcn_rcpf(rjv * rk);
        const float ssum   = rjsq + rksq + rjk_sq;

        float w = 0.5f * fjv * fk * fc_cut(rjk);   // fcprod * masks * 0.5
        w = (j == k) ? 0.0f : w;

        const float cm  = fmaxf(1.0f - cosv, 0.0f);
        const float cp  = fmaxf(1.0f + cosv, 0.0f);
        const float cm2 = cm * cm,  cp2 = cp * cp;
        const float cm4 = cm2 * cm2, cp4 = cp2 * cp2;
        float cp16 = cp4 * cp4; cp16 = cp16 * cp16;

#pragma unroll
        for (int e = 0; e < 7; ++e) {
          const float ef = exp2_raw(ETA7_L2[e] * ssum) * w;
          acc[6 * e + 0] += cm * ef;                 // zeta=1, lam=-1
          acc[6 * e + 1] += cp * ef;                 // zeta=1, lam=+1
          acc[6 * e + 2] += 0.5f   * cm2 * ef;       // zeta=2, lam=-1
          acc[6 * e + 3] += 0.5f   * cp2 * ef;       // zeta=2, lam=+1
          acc[6 * e + 4] += 0.125f * cm4 * ef;       // zeta=4, lam=-1
          acc[6 * e + 5] += 0.125f * cp4 * ef;       // zeta=4, lam=+1
          if (e == 6)                                 // eta=0.08, zeta=16
            acc[42] += 3.0517578125e-05f * cp16 * ef; // 2^(1-16)
        }
      }
    }
  }

  // ---- LDS transpose reduction ----
#pragma unroll
  for (int e = 0; e < N_G2; ++e) red[wv][e][lane] = g2v[e];
#pragma unroll
  for (int p = 0; p < N_G4; ++p) red[wv][N_G2 + p][lane] = acc[p];

  wave_lds_fence();

  // Lane l reduces feature l (and feature l+32): 8 independent b128 loads
  // each, pipelined by the LDS unit; rows are 16B-aligned (stride 144B).
  float* orow = out + (size_t)atom * N_FEAT;

  {
    const float4* row = (const float4*)&red[wv][lane][0];
    float4 a0 = row[0], a1 = row[1], a2 = row[2], a3 = row[3];
    float4 a4 = row[4], a5 = row[5], a6 = row[6], a7 = row[7];
    float s = ((a0.x + a0.y) + (a0.z + a0.w)) + ((a1.x + a1.y) + (a1.z + a1.w))
            + ((a2.x + a2.y) + (a2.z + a2.w)) + ((a3.x + a3.y) + (a3.z + a3.w))
            + ((a4.x + a4.y) + (a4.z + a4.w)) + ((a5.x + a5.y) + (a5.z + a5.w))
            + ((a6.x + a6.y) + (a6.z + a6.w)) + ((a7.x + a7.y) + (a7.z + a7.w));
    orow[lane] = s;                            // coalesced store, features 0..31
  }
  if (lane < N_FEAT - 32) {                    // features 32..50
    const float4* row = (const float4*)&red[wv][32 + lane][0];
    float4 a0 = row[0], a1 = row[1], a2 = row[2], a3 = row[3];
    float4 a4 = row[4], a5 = row[5], a6 = row[6], a7 = row[7];
    float s = ((a0.x + a0.y) + (a0.z + a0.w)) + ((a1.x + a1.y) + (a1.z + a1.w))
            + ((a2.x + a2.y) + (a2.z + a2.w)) + ((a3.x + a3.y) + (a3.z + a3.w))
            + ((a4.x + a4.y) + (a4.z + a4.w)) + ((a5.x + a5.y) + (a5.z + a5.w))
            + ((a6.x + a6.y) + (a6.z + a6.w)) + ((a7.x + a7.y) + (a7.z + a7.w));
    orow[32 + lane] = s;                       // coalesced store
  }
}

extern "C" void kernel_launch(void* const* d_in, const int* in_sizes, int n_in,
                              void* d_out, int out_size, void* d_ws, size_t ws_size,
                              hipStream_t stream) {
  (void)n_in; (void)out_size; (void)d_ws; (void)ws_size;
  const float* pos    = (const float*)d_in[0];
  const int*   numnei = (const int*)d_in[1];
  const int*   neighs = (const int*)d_in[2];
  float*       out    = (float*)d_out;
  const int n_atoms = in_sizes[1];           // len(numnei)
  const int blocks  = (n_atoms + 7) / 8;     // 8 atoms (waves) per 256-thread block
  behler_g2g4_kernel<<<blocks, 256, 0, stream>>>(pos, numnei, neighs, out, n_atoms);
}